// EHA_Circuit_69973607186816
// MI455X (gfx1250) — compile-verified
//
#include <hip/hip_runtime.h>

typedef __attribute__((ext_vector_type(2))) float v2f;
typedef __attribute__((ext_vector_type(8))) float v8f;

#define N_LAYERS 4
#define N_GROUPS 5
#define PPL      93
#define BATCH    64
#define D_STATE  65536
#define TOTAL_AMPS (BATCH * D_STATE)          // 4,194,304 complex
#define TOTAL_COLS (BATCH * (D_STATE / 16))   // 262,144 columns of 16
#define TILES      (TOTAL_COLS / 16)          // 16,384 wave-tiles

__device__ __forceinline__ float2 cmul(float2 a, float2 b) {
    return make_float2(a.x * b.x - a.y * b.y, a.x * b.y + a.y * b.x);
}
__device__ __forceinline__ float2 cadd(float2 a, float2 b) {
    return make_float2(a.x + b.x, a.y + b.y);
}

// ---------------------------------------------------------------------------
// Build the 20 fused 16x16 complex gate matrices (4 layers x 5 groups).
// Group g of a layer covers wires 3g..3g+3 and fuses:
//   singles: wires {0,1,2,3} for g==0, else {3g+1, 3g+2, 3g+3}
//   pairs  : (3g,3g+1), (3g+1,3g+2), (3g+2,3g+3)   in that order
// Layer 0 / group 4 additionally folds RY(ancilla_rot) on wires 14,15
// (applied BEFORE the layer singles, i.e. right-multiplied).
// Wire ordering: wire w <-> bit (15-w) of the state index (PennyLane MSB
// convention), so local wire k of a group is bit (3-k) of the 4-bit gate index.
// ---------------------------------------------------------------------------
__global__ __launch_bounds__(256)
void build_gates_kernel(const float* __restrict__ theta,
                        const float* __restrict__ anc_rot,
                        float2* __restrict__ gates) {
    const int blk = blockIdx.x;             // l*5 + g
    const int l   = blk / N_GROUPS;
    const int g   = blk % N_GROUPS;
    const int tid = threadIdx.x;            // 256 threads: one matrix element
    const int i   = tid >> 4;               // output row
    const int j   = tid & 15;               // input col
    const float* pp = theta + l * PPL;

    __shared__ float2 cur[256];

    // ---- Kronecker product of the (fused) single-qubit gates -------------
    float2 acc = make_float2(1.f, 0.f);
    for (int k = 0; k < 4; ++k) {
        const int w  = 3 * g + k;
        const int bi = (i >> (3 - k)) & 1;
        const int bj = (j >> (3 - k)) & 1;
        float2 Sv;
        const bool hasSingle = (g == 0) || (k > 0);
        if (!hasSingle) {
            Sv = (bi == bj) ? make_float2(1.f, 0.f) : make_float2(0.f, 0.f);
        } else {
            const float ax = pp[3 * w], ay = pp[3 * w + 1], az = pp[3 * w + 2];
            const float cx = cosf(0.5f * ax), sx = sinf(0.5f * ax);
            const float cy = cosf(0.5f * ay), sy = sinf(0.5f * ay);
            const float cz = cosf(0.5f * az), sz = sinf(0.5f * az);
            // RY*RX rows: r0 = [cy*cx + i*sy*sx , -sy*cx - i*cy*sx]
            //             r1 = [sy*cx - i*cy*sx ,  cy*cx - i*sy*sx]
            // then RZ scales row0 by e^{-i az/2}, row1 by e^{+i az/2}
            const float2 e0 = make_float2(cz, -sz);
            const float2 e1 = make_float2(cz,  sz);
            float2 U00 = cmul(e0, make_float2( cy * cx,  sy * sx));
            float2 U01 = cmul(e0, make_float2(-sy * cx, -cy * sx));
            float2 U10 = cmul(e1, make_float2( sy * cx, -cy * sx));
            float2 U11 = cmul(e1, make_float2( cy * cx, -sy * sx));
            if (l == 0 && w >= 14) {
                // fold the initial ancilla RY: U' = U @ RY(phi)
                const float phi = anc_rot[0];
                const float c = cosf(0.5f * phi), s = sinf(0.5f * phi);
                float2 n00 = make_float2(U00.x * c + U01.x * s, U00.y * c + U01.y * s);
                float2 n01 = make_float2(-U00.x * s + U01.x * c, -U00.y * s + U01.y * c);
                float2 n10 = make_float2(U10.x * c + U11.x * s, U10.y * c + U11.y * s);
                float2 n11 = make_float2(-U10.x * s + U11.x * c, -U10.y * s + U11.y * c);
                U00 = n00; U01 = n01; U10 = n10; U11 = n11;
            }
            Sv = bi ? (bj ? U11 : U10) : (bj ? U01 : U00);
        }
        acc = cmul(acc, Sv);
    }
    cur[tid] = acc;
    __syncthreads();

    // ---- three embedded 4x4 Ising pair gates: cur <- E_t * cur -----------
    for (int t = 0; t < 3; ++t) {
        const int w  = 3 * g + t;                       // pair (w, w+1)
        const float ax = pp[48 + 3 * w];
        const float ay = pp[48 + 3 * w + 1];
        const float az = pp[48 + 3 * w + 2];
        const float cx = cosf(0.5f * ax), sx = sinf(0.5f * ax);
        const float cy = cosf(0.5f * ay), sy = sinf(0.5f * ay);
        const float cz = cosf(0.5f * az), sz = sinf(0.5f * az);
        // ZZ*YY*XX nonzeros:
        //  [0][0]=[3][3]=d0*A,  [0][3]=[3][0]=d0*(i*B)
        //  [1][1]=[2][2]=d1*C,  [1][2]=[2][1]=d1*(-i*D)
        const float A_ = cy * cx + sy * sx;
        const float B_ = sy * cx - cy * sx;
        const float C_ = cy * cx - sy * sx;
        const float D_ = sy * cx + cy * sx;
        const float2 d0 = make_float2(cz, -sz);
        const float2 d1 = make_float2(cz,  sz);

        const int sh = 2 - t;                 // LSB of the 2-bit sub-index
        const int si = (i >> sh) & 3;
        float2 row[4];
        row[0] = row[1] = row[2] = row[3] = make_float2(0.f, 0.f);
        if (si == 0)      { row[0] = cmul(d0, make_float2(A_, 0.f)); row[3] = cmul(d0, make_float2(0.f, B_)); }
        else if (si == 3) { row[3] = cmul(d0, make_float2(A_, 0.f)); row[0] = cmul(d0, make_float2(0.f, B_)); }
        else if (si == 1) { row[1] = cmul(d1, make_float2(C_, 0.f)); row[2] = cmul(d1, make_float2(0.f, -D_)); }
        else              { row[2] = cmul(d1, make_float2(C_, 0.f)); row[1] = cmul(d1, make_float2(0.f, -D_)); }

        const int ibase = i & ~(3 << sh);
        float2 nv = make_float2(0.f, 0.f);
        for (int sm = 0; sm < 4; ++sm) {
            const int m = ibase | (sm << sh);
            nv = cadd(nv, cmul(row[sm], cur[(m << 4) | j]));
        }
        __syncthreads();
        cur[tid] = nv;
        __syncthreads();
    }

    gates[blk * 256 + tid] = cur[tid];
}

// ---------------------------------------------------------------------------
// Pack separate real/imag inputs into an interleaved complex state (in d_out).
// ---------------------------------------------------------------------------
__global__ __launch_bounds__(256)
void pack_state_kernel(const float* __restrict__ xr,
                       const float* __restrict__ xi,
                       float2* __restrict__ st) {
    const int n = blockIdx.x * 256 + threadIdx.x;
    st[n] = make_float2(xr[n], xi[n]);
}

// ---------------------------------------------------------------------------
// Apply one fused 16x16 complex gate to the whole batched state, in place.
// State index: n = hi*2^(p+4) + j*2^p + lo  (j = 4-bit gate axis, p = 12-3g).
// Column c = b*4096 + hi*2^p + lo; each wave owns one 16-column tile:
//   Out[16x16] = G[16x16] @ S[16x16]   (complex, via 16 V_WMMA_F32_16X16X4_F32)
// Read set == write set per wave -> in-place safe, no cross-wave overlap.
// f32 WMMA fragment layouts (ISA 7.12.2):
//   A 16x4 : lane l, vgpr v -> M = l%16, K = v + 2*(l/16)
//   B 4x16 : lane l, vgpr v -> N = l%16, K = v + 2*(l/16)
//   C/D    : lane l, vgpr r -> N = l%16, M = r + 8*(l/16)
// ---------------------------------------------------------------------------
__device__ __forceinline__ v8f wmma_f32(v2f a, v2f b, v8f c) {
    return __builtin_amdgcn_wmma_f32_16x16x4_f32(
        /*neg_a=*/false, a, /*neg_b=*/false, b,
        /*c_mod=*/(short)0, c, /*reuse_a=*/false, /*reuse_b=*/false);
}

__global__ __launch_bounds__(256)
void apply_group_kernel(float2* __restrict__ state,
                        const float2* __restrict__ G,
                        int p) {
    const int tid  = threadIdx.x;
    const int lane = tid & 31;
    const int wave = tid >> 5;                  // 0..7
    const int tile = blockIdx.x * 8 + wave;     // 0..16383
    const int n15  = lane & 15;
    const int lh   = lane >> 4;

    // ---- A fragments (gate matrix), imag pre-negated for the Re-part -----
    v2f Ar[4], Ai[4], AiN[4];
    #pragma unroll
    for (int t = 0; t < 4; ++t) {
        #pragma unroll
        for (int v = 0; v < 2; ++v) {
            const int j = 4 * t + v + 2 * lh;
            const float2 gv = G[n15 * 16 + j];
            Ar[t][v]  = gv.x;
            Ai[t][v]  = gv.y;
            AiN[t][v] = -gv.y;
        }
    }

    // ---- this lane's column -> base amplitude index -----------------------
    const int c0   = tile * 16 + n15;
    const int b    = c0 >> 12;
    const int r12  = c0 & 4095;
    const int hi   = r12 >> p;
    const int lo   = r12 & ((1 << p) - 1);
    const int base = (b << 16) + (hi << (p + 4)) + lo;   // j == 0 element

    // ---- load B tiles and accumulate complex matmul -----------------------
    v8f Dr = {}; v8f Di = {};
    #pragma unroll
    for (int t = 0; t < 4; ++t) {
        v2f Br, Bi;
        #pragma unroll
        for (int v = 0; v < 2; ++v) {
            const int j = 4 * t + v + 2 * lh;
            const float2 sv = state[base + (j << p)];
            Br[v] = sv.x;
            Bi[v] = sv.y;
        }
        Dr = wmma_f32(Ar[t],  Br, Dr);   // + Ur*Sr
        Di = wmma_f32(Ar[t],  Bi, Di);   // + Ur*Si
        Dr = wmma_f32(AiN[t], Bi, Dr);   // - Ui*Si
        Di = wmma_f32(Ai[t],  Br, Di);   // + Ui*Sr
    }

    // ---- store D tile back (in place) -------------------------------------
    #pragma unroll
    for (int r = 0; r < 8; ++r) {
        const int irow = r + 8 * lh;
        state[base + (irow << p)] = make_float2(Dr[r], Di[r]);
    }
}

// ---------------------------------------------------------------------------
extern "C" void kernel_launch(void* const* d_in, const int* in_sizes, int n_in,
                              void* d_out, int out_size, void* d_ws, size_t ws_size,
                              hipStream_t stream) {
    (void)in_sizes; (void)n_in; (void)out_size; (void)ws_size;

    const float* xr    = (const float*)d_in[0];   // [64, 65536]
    const float* xi    = (const float*)d_in[1];   // [64, 65536]
    const float* theta = (const float*)d_in[2];   // [4, 93]
    const float* anc   = (const float*)d_in[3];   // scalar

    float2* state = (float2*)d_out;               // [64, 65536] complex64
    float2* gates = (float2*)d_ws;                // 20 x 16 x 16 complex (40 KB)

    build_gates_kernel<<<N_LAYERS * N_GROUPS, 256, 0, stream>>>(theta, anc, gates);
    pack_state_kernel<<<TOTAL_AMPS / 256, 256, 0, stream>>>(xr, xi, state);

    for (int l = 0; l < N_LAYERS; ++l) {
        for (int g = 0; g < N_GROUPS; ++g) {
            apply_group_kernel<<<TILES / 8, 256, 0, stream>>>(
                state, gates + (l * N_GROUPS + g) * 256, 12 - 3 * g);
        }
    }
}